// RandomFeatureMasker_20598663151934
// MI455X (gfx1250) — compile-verified
//
#include <hip/hip_runtime.h>
#include <hip/hip_bf16.h>
#include <stdint.h>

// ---------------------------------------------------------------------------
// RandomFeatureMasker for MI455X (gfx1250, wave32).
// One wave per row (F=128 -> 4 floats/lane). Streaming kernel, ~1 GB traffic,
// HBM-roofline ~44 us @ 23.3 TB/s. Async global->LDS staging (double
// buffered, ASYNCcnt) + NT stores; selection via wave32 ds_bpermute
// butterflies.
// ---------------------------------------------------------------------------

typedef float vf4 __attribute__((ext_vector_type(4)));
typedef int v4i __attribute__((vector_size(16)));                 // matches builtin param
typedef __attribute__((address_space(1))) v4i* gv4i_p;            // global int4*
typedef __attribute__((address_space(3))) v4i* lv4i_p;            // LDS int4*

#define F_FEAT 128
#define WPB 8 // waves per block (256 threads)

#if __has_builtin(__builtin_amdgcn_global_load_async_to_lds_b128)
#define HAVE_ASYNC_LDS 1
#else
#define HAVE_ASYNC_LDS 0
#endif

__device__ __forceinline__ void wait_async_0() {
#if __has_builtin(__builtin_amdgcn_s_wait_asynccnt)
  __builtin_amdgcn_s_wait_asynccnt(0);
#else
  __asm__ __volatile__("s_wait_asynccnt 0" ::: "memory");
#endif
}
__device__ __forceinline__ void wait_async_2() {
#if __has_builtin(__builtin_amdgcn_s_wait_asynccnt)
  __builtin_amdgcn_s_wait_asynccnt(2);
#else
  __asm__ __volatile__("s_wait_asynccnt 2" ::: "memory");
#endif
}

#if HAVE_ASYNC_LDS
// Per-lane 16B async copy global -> LDS (GLOBAL_LOAD_ASYNC_TO_LDS_B128).
// Pointers built via integer round-trip: generic LDS pointers carry the LDS
// offset in the low 32 bits on AMDGPU; global generic == AS1 bit pattern.
__device__ __forceinline__ void async_cp_b128(const float* gsrc, float* ldst) {
  unsigned long long g = (unsigned long long)(uintptr_t)gsrc;
  unsigned l = (unsigned)(uintptr_t)ldst;
  __builtin_amdgcn_global_load_async_to_lds_b128((gv4i_p)g, (lv4i_p)l, 0, 0);
}
#endif

__device__ __forceinline__ unsigned pcg(unsigned v) {
  v ^= v >> 16; v *= 0x7feb352du;
  v ^= v >> 15; v *= 0x846ca68bu;
  v ^= v >> 16;
  return v;
}
__device__ __forceinline__ float u01(unsigned h) {
  return (float)(h >> 8) * (1.0f / 16777216.0f);
}

__global__ __launch_bounds__(256) void RandomFeatureMasker_20598663151934_kernel(
    const float* __restrict__ x, const float* __restrict__ mask,
    float* __restrict__ outx, float* __restrict__ outm, int nrows) {
  __shared__ __align__(16) float bx[WPB][2][F_FEAT];
  __shared__ __align__(16) float bm[WPB][2][F_FEAT];

  const int lane = threadIdx.x & 31;
  const int wslot = threadIdx.x >> 5;
  const long long stride = (long long)gridDim.x * WPB;
  const long long w0 = (long long)blockIdx.x * WPB + wslot;
  const int l4 = lane * 4;

  int buf = 0;
#if HAVE_ASYNC_LDS
  if (w0 < nrows) {
    async_cp_b128(x + (size_t)w0 * F_FEAT + l4, &bx[wslot][0][l4]);
    async_cp_b128(mask + (size_t)w0 * F_FEAT + l4, &bm[wslot][0][l4]);
  }
#endif

  for (long long r = w0; r < nrows; r += stride) {
    vf4 xv, mv;
#if HAVE_ASYNC_LDS
    long long rn = r + stride;
    if (rn < nrows) { // prefetch next row into the other buffer
      async_cp_b128(x + (size_t)rn * F_FEAT + l4, &bx[wslot][buf ^ 1][l4]);
      async_cp_b128(mask + (size_t)rn * F_FEAT + l4, &bm[wslot][buf ^ 1][l4]);
      wait_async_2(); // current row's 2 copies done, next row's in flight
    } else {
      wait_async_0();
    }
    __asm__ __volatile__("" ::: "memory");
    xv = *(const vf4*)&bx[wslot][buf][l4];
    mv = *(const vf4*)&bm[wslot][buf][l4];
#else
    xv = __builtin_nontemporal_load((const vf4*)(x + (size_t)r * F_FEAT + l4));
    mv = __builtin_nontemporal_load((const vf4*)(mask + (size_t)r * F_FEAT + l4));
#endif

    const unsigned row = (unsigned)r;
    float xe[4], me[4], s[4];
    int vcnt = 0;
#pragma unroll
    for (int j = 0; j < 4; ++j) {
      xe[j] = xv[j];
      me[j] = mv[j];
      const bool valid = mv[j] > 0.5f;
      vcnt += valid ? 1 : 0;
      const unsigned col = (unsigned)(l4 + j);
      s[j] = valid ? u01(pcg(pcg(row) ^ (col * 0x9E3779B9u))) : -1.0f;
    }

    // n_valid across the wave (wave32 butterfly -> ds_bpermute)
    int nv = vcnt;
#pragma unroll
    for (int off = 16; off > 0; off >>= 1) nv += __shfl_xor(nv, off, 32);

    // Poisson(2) by CDF inversion; wave-uniform (same row key in all lanes)
    const float up = u01(pcg(row ^ 0xB5297A4Du));
    int n = 0;
    float p = 0.13533528f; // e^-2
    float c = p;
    while (c < up && n < F_FEAT - 1) {
      ++n;
      p *= 2.0f / (float)n;
      c += p;
    }
    n = (nv > 1) ? (n < nv - 1 ? n : nv - 1) : 0;

    // Drop the n highest-scoring valid positions: iterative wave-max extract.
    for (int k = 0; k < n; ++k) {
      float best = s[0];
      int gidx = l4;
#pragma unroll
      for (int j = 1; j < 4; ++j)
        if (s[j] > best) { best = s[j]; gidx = l4 + j; }
#pragma unroll
      for (int off = 16; off > 0; off >>= 1) {
        float ov = __shfl_xor(best, off, 32);
        int oi = __shfl_xor(gidx, off, 32);
        if (ov > best || (ov == best && oi < gidx)) { best = ov; gidx = oi; }
      }
#pragma unroll
      for (int j = 0; j < 4; ++j) {
        if (gidx == l4 + j) { s[j] = -1.0f; xe[j] = 0.0f; me[j] = 0.0f; }
      }
    }

    vf4 ox = {xe[0], xe[1], xe[2], xe[3]};
    vf4 om = {me[0], me[1], me[2], me[3]};
    __builtin_nontemporal_store(ox, (vf4*)(outx + (size_t)r * F_FEAT + l4));
    __builtin_nontemporal_store(om, (vf4*)(outm + (size_t)r * F_FEAT + l4));
    buf ^= 1;
  }
}

extern "C" void kernel_launch(void* const* d_in, const int* in_sizes, int n_in,
                              void* d_out, int out_size, void* d_ws, size_t ws_size,
                              hipStream_t stream) {
  const float* x = (const float*)d_in[0];
  const float* mask = (const float*)d_in[1];
  float* out = (float*)d_out;

  const int nrows = in_sizes[0] / F_FEAT; // 500000
  float* outx = out;
  float* outm = out + (size_t)nrows * F_FEAT;

  int blocks = (nrows + WPB - 1) / WPB;
  if (blocks > 4096) blocks = 4096; // grid-stride, 32K waves resident
  RandomFeatureMasker_20598663151934_kernel<<<blocks, 256, 0, stream>>>(
      x, mask, outx, outm, nrows);
}